// MultiGraphAttention_47184510713875
// MI455X (gfx1250) — compile-verified
//
#include <hip/hip_runtime.h>

// ---------------------------------------------------------------------------
// MultiGraphAttention on gfx1250 (MI455X, wave32)
//   B=8, N=4096, F=3, H=4, C=16, K=16
//   Kernel 1: exact 16-NN — selection-bound; candidate tiles staged with
//             GLOBAL_LOAD_ASYNC_TO_LDS_B32 (double-buffered, ASYNCcnt).
//   Kernel 2: per-point pc-branch score pc2[b,n,h] (tiny MLP).
//   Kernel 3: fused neighbor-MLP (V_WMMA_F32_16X16X4_F32, F padded 3->4) +
//             leaky-softmax attention over K in WMMA register layout,
//             cross-lane reductions via immediate ds_swizzle.
// ---------------------------------------------------------------------------

typedef __attribute__((ext_vector_type(2))) float v2f;
typedef __attribute__((ext_vector_type(8))) float v8f;

#define NB   8
#define NPTS 4096
#define NF   3
#define NH   4
#define NC   16
#define NK   16

// xor-lane swizzle: group-of-32 mode, result_lane = lane ^ M
template <int M>
__device__ __forceinline__ float swz_xor(float x) {
  constexpr int imm = (M << 10) | 0x1F;   // and=0x1F, or=0, xor=M, offset[15]=0
  return __int_as_float(__builtin_amdgcn_ds_swizzle(__float_as_int(x), imm));
}

// ---------------- Kernel 1: exact 16-NN, one query per thread --------------
#define KNN_TILE 128   // 128 points * 12 B = 1536 B per tile

// Async memory->LDS tile copy (GVS mode: vdst = LDS byte addr, vaddr = 32-bit
// per-lane offset, saddr = uniform 64-bit base).  Per ISA 08 pseudocode the
// instruction offset is added to BOTH the LDS and the global address, so three
// instructions at offset 0/512/1024 move the whole 1536-B tile (128 thr x 4B).
__device__ __forceinline__ void async_tile_copy(uint32_t lds_addr, uint32_t voff,
                                                const void* sbase) {
  asm volatile(
      "global_load_async_to_lds_b32 %0, %1, %2 offset:0\n"
      "global_load_async_to_lds_b32 %0, %1, %2 offset:512\n"
      "global_load_async_to_lds_b32 %0, %1, %2 offset:1024"
      :: "v"(lds_addr), "v"(voff), "s"(sbase)
      : "memory");
}

__global__ __launch_bounds__(KNN_TILE)
void knn_kernel(const float* __restrict__ pc, int* __restrict__ nn_idx) {
  __shared__ float sh[2][KNN_TILE * 3];
  const int tid = threadIdx.x;
  const int q   = blockIdx.x * KNN_TILE + tid;          // global query (b*N+n)
  const int b   = blockIdx.x / (NPTS / KNN_TILE);       // uniform per block

  const float* qp = pc + (size_t)q * 3;
  const float qx = qp[0], qy = qp[1], qz = qp[2];

  // sorted ascending by (dist, index); ties keep lower index (scan order)
  float bd[NK];
  int   bi[NK];
#pragma unroll
  for (int t = 0; t < NK; ++t) { bd[t] = 3.4e38f; bi[t] = 0; }

  const float* base = pc + (size_t)b * NPTS * 3;
  const uint32_t voff = (uint32_t)tid * 4u;
  const uint32_t lds0 = (uint32_t)(uintptr_t)(&sh[0][0]) + voff;
  const uint32_t lds1 = (uint32_t)(uintptr_t)(&sh[1][0]) + voff;
  constexpr int NT = NPTS / KNN_TILE;                   // 32 tiles

  async_tile_copy(lds0, voff, (const void*)base);       // prologue: tile 0
  for (int tile = 0; tile < NT; ++tile) {
    const int cur = tile & 1;
    if (tile + 1 < NT) {
      // prefetch next tile into the other buffer, then wait for current
      async_tile_copy(cur ? lds0 : lds1, voff,
                      (const void*)(base + (size_t)(tile + 1) * KNN_TILE * 3));
      asm volatile("s_wait_asynccnt 0x3" ::: "memory");  // 3 newest in flight
    } else {
      asm volatile("s_wait_asynccnt 0x0" ::: "memory");
    }
    __syncthreads();                                     // cur buffer visible

    const float* s = sh[cur];
#pragma unroll 4
    for (int j = 0; j < KNN_TILE; ++j) {
      const float dx = qx - s[j * 3 + 0];
      const float dy = qy - s[j * 3 + 1];
      const float dz = qz - s[j * 3 + 2];
      const float d  = dx * dx + dy * dy + dz * dz;
      if (d < bd[NK - 1]) {                              // rare after warm-up
        const int idx = tile * KNN_TILE + j;
        int rank = 0;
#pragma unroll
        for (int t = 0; t < NK; ++t) rank += (bd[t] <= d) ? 1 : 0;
#pragma unroll
        for (int t = NK - 1; t > 0; --t) {
          if (t > rank) { bd[t] = bd[t - 1]; bi[t] = bi[t - 1]; }
        }
#pragma unroll
        for (int t = 0; t < NK; ++t) {
          if (t == rank) { bd[t] = d; bi[t] = idx; }
        }
      }
    }
    __syncthreads();       // all reads of cur done before it is refilled
  }
#pragma unroll
  for (int t = 0; t < NK; ++t) nn_idx[(size_t)q * NK + t] = bi[t];
}

// ---------------- Kernel 2: pc2[b,n,h] = relu(relu(p*W1+b1)*W2+b2) ---------
__global__ __launch_bounds__(256)
void pc_score_kernel(const float* __restrict__ pc,
                     const float* __restrict__ w1, const float* __restrict__ b1,
                     const float* __restrict__ w2, const float* __restrict__ b2,
                     float* __restrict__ s2) {
  const int id = blockIdx.x * 256 + threadIdx.x;   // [B*N*H]
  const int h  = id & (NH - 1);
  const int bn = id >> 2;
  const float* p = pc + (size_t)bn * 3;
  const float x = p[0], y = p[1], z = p[2];
  const float* w1h = w1 + h * NF * NC;   // [F,C]
  const float* w2h = w2 + h * NC;        // [C]
  float acc = b2[h];
#pragma unroll
  for (int c = 0; c < NC; ++c) {
    float t = x * w1h[c] + y * w1h[NC + c] + z * w1h[2 * NC + c] + b1[h * NC + c];
    t = fmaxf(t, 0.f);
    acc += t * w2h[c];
  }
  s2[id] = fmaxf(acc, 0.f);
}

// ---------------- Kernel 3: fused WMMA neighbor-MLP + attention ------------
// One wave per (b,n). A = 16 gathered neighbors x 4 feats (padded),
// B(head h) = 4 x 16 slice of w_knn1, D = gf[h] 16x16 in 8 VGPRs.
// WMMA f32 16x16x4 layouts (ISA 7.12.2):
//   A: lane<16 -> row=lane, VGPR0=K0,VGPR1=K1 ; lane>=16 -> VGPR0=K2,VGPR1=K3
//   B: col=lane&15,         lane<16: VGPR0=K0,VGPR1=K1 ; lane>=16: K2,K3
//   D: lane<16: VGPRv = row v, col=lane ; lane>=16: row v+8, col=lane-16
__global__ __launch_bounds__(256)
void attn_kernel(const float* __restrict__ pc,
                 const int*   __restrict__ nn_idx,
                 const float* __restrict__ s2,
                 const float* __restrict__ w1, const float* __restrict__ b1,
                 const float* __restrict__ w2, const float* __restrict__ b2,
                 float* __restrict__ out_att,   // [B,N,H,C]
                 float* __restrict__ out_gf,    // [B,N,H,K,C]
                 float* __restrict__ out_cf) {  // [B,N,H,K]
  const int lane = threadIdx.x & 31;
  const int bn   = (blockIdx.x * blockDim.x + threadIdx.x) >> 5;
  const int b    = bn >> 12;            // / NPTS
  const bool hiw = lane >= 16;
  const int  kc  = lane & 15;           // neighbor row (A) / column (B,D)
  const int  krow0 = hiw ? 8 : 0;

  // ---- gather neighbor point -> A matrix (F padded to 4 with zeros) ----
  const int nb = nn_idx[(size_t)bn * NK + kc];
  const float* p = pc + ((size_t)b * NPTS + nb) * 3;
  const float px = p[0], py = p[1], pz = p[2];
  v2f A;
  A.x = hiw ? pz : px;    // K2 : K0
  A.y = hiw ? 0.f : py;   // K3(pad) : K1

  float* gout = out_gf  + (size_t)bn * NH * NK * NC;
  float* aout = out_att + (size_t)bn * NH * NC;
  float* cout = out_cf  + (size_t)bn * NH * NK;

#pragma unroll
  for (int h = 0; h < NH; ++h) {
    const float* w1h = w1 + h * NF * NC;        // [F=3, C=16]
    v2f Bm;
    Bm.x = hiw ? w1h[2 * NC + kc] : w1h[kc];    // K2 : K0
    Bm.y = hiw ? 0.f : w1h[NC + kc];            // K3(pad) : K1

    v8f acc = {};
    // 8-arg form: (neg_a, A, neg_b, B, c_mod, C, reuse_a, reuse_b)
    acc = __builtin_amdgcn_wmma_f32_16x16x4_f32(
        false, A, false, Bm, (short)0, acc, false, false);

    // ---- bias + relu -> graph features gf[h][k = v+krow0][c = kc] ----
    const float bias = b1[h * NC + kc];
    float gf[8];
#pragma unroll
    for (int v = 0; v < 8; ++v) gf[v] = fmaxf(acc[v] + bias, 0.f);

    // store multi_graph_features [b,n,h,k,c]
#pragma unroll
    for (int v = 0; v < 8; ++v)
      gout[h * NK * NC + (krow0 + v) * NC + kc] = gf[v];

    // ---- kn2[k] = relu(sum_c gf[k,c]*w2[h,c] + b2[h]) : lane-reduce cols --
    const float w2c = w2[h * NC + kc];
    const float b2h = b2[h];
    const float s2h = s2[bn * NH + h];
    float lv[8];
#pragma unroll
    for (int v = 0; v < 8; ++v) {
      float r = gf[v] * w2c;
      r += swz_xor<1>(r);
      r += swz_xor<2>(r);
      r += swz_xor<4>(r);
      r += swz_xor<8>(r);
      lv[v] = r;                      // kn2 pre-bias, all 16 lanes of half
    }
    // logits = pc2 + kn2, then LeakyReLU(0.3)
#pragma unroll
    for (int v = 0; v < 8; ++v) {
      const float kn2 = fmaxf(lv[v] + b2h, 0.f);
      const float lg  = s2h + kn2;
      lv[v] = (lg >= 0.f) ? lg : 0.3f * lg;
    }

    // ---- softmax over 16 k values (8 regs x 2 half-waves) ----
    float mx = lv[0];
#pragma unroll
    for (int v = 1; v < 8; ++v) mx = fmaxf(mx, lv[v]);
    mx = fmaxf(mx, swz_xor<16>(mx));
    float sum = 0.f;
#pragma unroll
    for (int v = 0; v < 8; ++v) { lv[v] = __expf(lv[v] - mx); sum += lv[v]; }
    sum += swz_xor<16>(sum);
    const float inv = 1.0f / sum;
#pragma unroll
    for (int v = 0; v < 8; ++v) lv[v] *= inv;

    // store multi_attention_coefficients [b,n,h,k] (one lane per half-wave)
    if (kc == 0) {
#pragma unroll
      for (int v = 0; v < 8; ++v) cout[h * NK + krow0 + v] = lv[v];
    }

    // ---- att[h,c] = sum_k coeff[k]*gf[k,c] ----
    float att = 0.f;
#pragma unroll
    for (int v = 0; v < 8; ++v) att += lv[v] * gf[v];
    att += swz_xor<16>(att);
    if (!hiw) aout[h * NC + kc] = att;   // lanes 0..15 hold c=0..15
  }
}

// ---------------------------------------------------------------------------
extern "C" void kernel_launch(void* const* d_in, const int* in_sizes, int n_in,
                              void* d_out, int out_size, void* d_ws, size_t ws_size,
                              hipStream_t stream) {
  const float* pc     = (const float*)d_in[0];
  const float* w_pc1  = (const float*)d_in[1];
  const float* b_pc1  = (const float*)d_in[2];
  const float* w_pc2  = (const float*)d_in[3];
  const float* b_pc2  = (const float*)d_in[4];
  const float* w_knn1 = (const float*)d_in[5];
  const float* b_knn1 = (const float*)d_in[6];
  const float* w_knn2 = (const float*)d_in[7];
  const float* b_knn2 = (const float*)d_in[8];

  // workspace: nn_idx [B,N,16] ints (2 MB), then s2 [B,N,H] floats (512 KB)
  int*   nn = (int*)d_ws;
  float* s2 = (float*)((char*)d_ws + (size_t)NB * NPTS * NK * sizeof(int));

  // outputs concatenated flat in return order
  float* out_att = (float*)d_out;                               // [B,N,H,C]
  float* out_gf  = out_att + (size_t)NB * NPTS * NH * NC;       // [B,N,H,K,C]
  float* out_cf  = out_gf  + (size_t)NB * NPTS * NH * NK * NC;  // [B,N,H,K]

  knn_kernel<<<NB * NPTS / KNN_TILE, KNN_TILE, 0, stream>>>(pc, nn);
  pc_score_kernel<<<NB * NPTS * NH / 256, 256, 0, stream>>>(
      pc, w_pc1, b_pc1, w_pc2, b_pc2, s2);
  attn_kernel<<<NB * NPTS / 8, 256, 0, stream>>>(
      pc, nn, s2, w_knn1, b_knn1, w_knn2, b_knn2, out_att, out_gf, out_cf);
}